// PairwiseProjectionModule_86921548136923
// MI455X (gfx1250) — compile-verified
//
#include <hip/hip_runtime.h>
#include <hip/hip_bf16.h>
#include <math.h>

typedef __attribute__((ext_vector_type(2))) float v2f;
typedef __attribute__((ext_vector_type(8))) float v8f;

#define Lr 256
#define Dd 128
#define Cc 32
#define Bb 4
#define NN (Lr * Lr)

// workspace layout (in floats)
#define FLAT_OFF 0
#define PART_STRIDE 1056                      // 32 ksum + 1024 kv per chunk
#define PART_OFF (Bb * NN * Cc)               // 8,388,608
#define FIN_OFF (PART_OFF + Bb * 256 * PART_STRIDE)

#define EPS_ATTN 1e-6f
#define EPS_LN   1e-5f

__device__ __forceinline__ float phi_f(float v) {
  // elu(v)+1
  return v > 0.0f ? v + 1.0f : __expf(v);
}

// ---------------------------------------------------------------------------
// Kernel 1: flat[b,i,j,c] = scale*sum_d w[c,d] x[b,i,d] x[b,j,d] + conv_b[c]
// Per block (b,i): GEMM  X[b] (256x128)  @  W_i^T (128x32)  via f32 WMMA.
// 8 waves x 4 tiles = 32 tiles of 16x16 (16 j-tiles x 2 c-tiles), K=128 in
// steps of 4 (V_WMMA_F32_16X16X4_F32).
// ---------------------------------------------------------------------------
__global__ __launch_bounds__(256)
void k1_pairproj(const float* __restrict__ x, const float* __restrict__ conv_w,
                 const float* __restrict__ conv_b, const float* __restrict__ scale,
                 float* __restrict__ flat) {
  const int i = blockIdx.x;
  const int b = blockIdx.y;
  const int tid = threadIdx.x;
  __shared__ float WiT[Cc][Dd];  // WiT[c][d] = scale * w[c,d] * x[b,i,d]  (B-matrix, transposed)

  const float s = scale[0];
  const float* xi = x + (size_t)(b * Lr + i) * Dd;
  for (int e = tid; e < Cc * Dd; e += 256) {
    const int c = e >> 7, d = e & 127;
    WiT[c][d] = s * conv_w[c * Dd + d] * xi[d];
  }
  __syncthreads();

  const int wave = tid >> 5;
  const int lane = tid & 31;
  const int lr = lane & 15;      // row (A) / col (B) within the 16x16 tile
  const int hi = lane >> 4;      // K-half select: lanes 16-31 hold K+2,K+3
  const float* xb = x + (size_t)b * Lr * Dd;
  float* outrow = flat + (size_t)(b * Lr + i) * Lr * Cc;

  for (int t = wave * 4; t < wave * 4 + 4; ++t) {
    const int jt = t >> 1;       // j tile 0..15
    const int ct = t & 1;        // c tile 0..1
    v8f acc = {0.f, 0.f, 0.f, 0.f, 0.f, 0.f, 0.f, 0.f};
    // A fragment source: x[b, jt*16+lr, k + 2*hi .. +1]   (row major, float2)
    const float* arow = xb + (size_t)(jt * 16 + lr) * Dd + hi * 2;
    // B fragment source: WiT[ct*16+lr][k + 2*hi .. +1]  => B[K][N=c] mirrored layout
    const float* brow = &WiT[ct * 16 + lr][hi * 2];
#pragma unroll
    for (int k = 0; k < Dd; k += 4) {
      v2f a = *(const v2f*)(arow + k);
      v2f bb = *(const v2f*)(brow + k);
      acc = __builtin_amdgcn_wmma_f32_16x16x4_f32(false, a, false, bb,
                                                  (short)0, acc, false, false);
    }
    const float cb = conv_b[ct * 16 + lr];  // N = c = ct*16 + lane%16, same for all 8 VGPRs
#pragma unroll
    for (int v = 0; v < 8; ++v) {
      const int j = jt * 16 + v + hi * 8;   // M = v (lanes 0-15) or v+8 (lanes 16-31)
      outrow[(size_t)j * Cc + ct * 16 + lr] = acc[v] + cb;
    }
  }
}

// ---------------------------------------------------------------------------
// Kernel 2: per-chunk partial reductions for linear attention.
// chunk = 256 consecutive n positions. partials: [32 ksum][1024 kv].
//   ksum[c]  = sum_n phi(f[n,c])
//   kv[d,e]  = sum_n phi(f[n,d]) * f[n,e]
// ---------------------------------------------------------------------------
__global__ __launch_bounds__(256)
void k2_partial(const float* __restrict__ flat, float* __restrict__ part) {
  const int chunk = blockIdx.x;
  const int b = blockIdx.y;
  const int tid = threadIdx.x;
  __shared__ float Fs[256 * Cc];  // 32 KB

  const float* src = flat + ((size_t)b * NN + (size_t)chunk * 256) * Cc;
  for (int e = tid; e < 256 * Cc; e += 256) Fs[e] = src[e];
  __syncthreads();

  float* pout = part + (size_t)(b * 256 + chunk) * PART_STRIDE;
  if (tid < Cc) {
    float acc = 0.f;
    for (int j = 0; j < 256; ++j) acc += phi_f(Fs[j * Cc + tid]);
    pout[tid] = acc;
  }
#pragma unroll
  for (int q = 0; q < 4; ++q) {
    const int ent = tid * 4 + q;        // 0..1023
    const int d = ent >> 5, e = ent & 31;
    float acc = 0.f;
    for (int j = 0; j < 256; ++j) acc += phi_f(Fs[j * Cc + d]) * Fs[j * Cc + e];
    pout[Cc + ent] = acc;
  }
}

// ---------------------------------------------------------------------------
// Kernel 3: deterministic final reduction of 256 chunk partials per batch.
// ---------------------------------------------------------------------------
__global__ __launch_bounds__(256)
void k3_finalreduce(const float* __restrict__ part, float* __restrict__ fin) {
  const int b = blockIdx.x;
  const int tid = threadIdx.x;
  const float* p = part + (size_t)b * 256 * PART_STRIDE;
#pragma unroll
  for (int q = 0; q < 4; ++q) {
    const int ent = tid + q * 256;      // kv entry 0..1023
    float acc = 0.f;
    for (int ch = 0; ch < 256; ++ch) acc += p[(size_t)ch * PART_STRIDE + Cc + ent];
    fin[(size_t)b * PART_STRIDE + Cc + ent] = acc;
  }
  if (tid < Cc) {
    float acc = 0.f;
    for (int ch = 0; ch < 256; ++ch) acc += p[(size_t)ch * PART_STRIDE + tid];
    fin[(size_t)b * PART_STRIDE + tid] = acc;
  }
}

// ---------------------------------------------------------------------------
// Kernel 4: attention normalize + residual + LayerNorm + symmetrize.
// Each thread owns (b,i,j); computes normalized value at (i,j) AND (j,i),
// writes the average. flat lives in workspace, out in d_out -> no races.
// ---------------------------------------------------------------------------
__device__ __forceinline__ void attn_ln_pos(const float* __restrict__ q,
                                            const float* __restrict__ kvs,
                                            const float* __restrict__ kss,
                                            const float* __restrict__ lw,
                                            const float* __restrict__ lb,
                                            float* __restrict__ outv) {
  float qv[Cc], qp[Cc];
  float z = EPS_ATTN;
#pragma unroll
  for (int d = 0; d < Cc; ++d) {
    qv[d] = q[d];
    qp[d] = phi_f(qv[d]);
    z += qp[d] * kss[d];
  }
  const float invz = 1.0f / z;
  float h[Cc];
  float mu = 0.f;
#pragma unroll
  for (int e = 0; e < Cc; ++e) {
    float ctx = 0.f;
#pragma unroll
    for (int d = 0; d < Cc; ++d) ctx += qp[d] * kvs[d * Cc + e];
    h[e] = qv[e] + ctx * invz;
    mu += h[e];
  }
  mu *= (1.0f / Cc);
  float var = 0.f;
#pragma unroll
  for (int e = 0; e < Cc; ++e) {
    const float dd = h[e] - mu;
    var += dd * dd;
  }
  var *= (1.0f / Cc);
  const float r = rsqrtf(var + EPS_LN);
#pragma unroll
  for (int e = 0; e < Cc; ++e) outv[e] = (h[e] - mu) * r * lw[e] + lb[e];
}

__global__ __launch_bounds__(256)
void k4_attn_ln_sym(const float* __restrict__ flat, const float* __restrict__ fin,
                    const float* __restrict__ ln_w, const float* __restrict__ ln_b,
                    float* __restrict__ out) {
  const int b = blockIdx.y;
  const int tid = threadIdx.x;
  __shared__ float kvs[Cc * Cc];
  __shared__ float kss[Cc];
  __shared__ float lw[Cc], lb[Cc];

  const float* f = fin + (size_t)b * PART_STRIDE;
  for (int e = tid; e < Cc * Cc; e += 256) kvs[e] = f[Cc + e];
  if (tid < Cc) {
    kss[tid] = f[tid];
    lw[tid] = ln_w[tid];
    lb[tid] = ln_b[tid];
  }
  __syncthreads();

  const int n = blockIdx.x * 256 + tid;
  const int i = n >> 8, j = n & 255;

  float a1[Cc], a2[Cc];
  attn_ln_pos(flat + ((size_t)b * NN + (size_t)i * Lr + j) * Cc, kvs, kss, lw, lb, a1);
  attn_ln_pos(flat + ((size_t)b * NN + (size_t)j * Lr + i) * Cc, kvs, kss, lw, lb, a2);

  float* o = out + ((size_t)b * NN + n) * Cc;
#pragma unroll
  for (int e = 0; e < Cc; ++e) o[e] = 0.5f * (a1[e] + a2[e]);
}

// ---------------------------------------------------------------------------
extern "C" void kernel_launch(void* const* d_in, const int* in_sizes, int n_in,
                              void* d_out, int out_size, void* d_ws, size_t ws_size,
                              hipStream_t stream) {
  const float* x      = (const float*)d_in[0];  // [4,256,128]
  const float* conv_w = (const float*)d_in[1];  // [32,128]
  const float* conv_b = (const float*)d_in[2];  // [32]
  const float* ln_w   = (const float*)d_in[3];  // [32]
  const float* ln_b   = (const float*)d_in[4];  // [32]
  const float* scale  = (const float*)d_in[5];  // [1]
  float* out = (float*)d_out;                   // [4,256,256,32]

  float* ws   = (float*)d_ws;
  float* flat = ws + FLAT_OFF;   // 33.5 MB
  float* part = ws + PART_OFF;   // 4.3 MB
  float* fin  = ws + FIN_OFF;    // 17 KB

  k1_pairproj<<<dim3(Lr, Bb), 256, 0, stream>>>(x, conv_w, conv_b, scale, flat);
  k2_partial<<<dim3(256, Bb), 256, 0, stream>>>(flat, part);
  k3_finalreduce<<<dim3(Bb), 256, 0, stream>>>(part, fin);
  k4_attn_ln_sym<<<dim3(256, Bb), 256, 0, stream>>>(flat, fin, ln_w, ln_b, out);
}